// SoftSuperpixelNeighborhoodAttention_16398185136403
// MI455X (gfx1250) — compile-verified
//
#include <hip/hip_runtime.h>
#include <math.h>

// Problem constants (from reference)
#define BB   2
#define HH   96
#define WW   96
#define DIMC 64
#define HEADS 2
#define HD   32
#define KS   7
#define K2   49
#define SHP  12
#define SWP  12
#define HWPIX (HH*WW)            // 9216
#define NPIX (BB*HWPIX)          // 18432
#define QKVW 192                 // q(64) | k(64) | v(64)
#define SCALE 0.17677669529663687f   // 32^-0.5
#define EPSV 1e-12f

typedef __attribute__((ext_vector_type(2))) float v2f;
typedef __attribute__((ext_vector_type(8))) float v8f;

// ---------------------------------------------------------------------------
// Kernel 1: QKV projection GEMM  (M=18432, K=64, N=192) with fp32 WMMA.
//   A[m][k] = x[b][k][i][j]   (x layout B,DIM,H,W)
//   B[k][n] = n<128 ? w_qk[k][n] : w_v[k][n-128]
// One wave per 16x16 output tile; 16 WMMA_F32_16X16X4_F32 steps over K.
// Block = 128 threads (4 waves) -> 4 N-tiles; grid.y=3 covers all 12 N-tiles.
// ---------------------------------------------------------------------------
__global__ void __launch_bounds__(128)
qkv_gemm_kernel(const float* __restrict__ x,
                const float* __restrict__ w_qk,
                const float* __restrict__ w_v,
                float* __restrict__ qkv) {
  const int lane = threadIdx.x & 31;
  const int wv   = threadIdx.x >> 5;            // 0..3
  const int m0   = blockIdx.x * 16;             // pixel tile base
  const int nt   = blockIdx.y * 4 + wv;         // 0..11
  const int n0   = nt * 16;

  // A fragment mapping (32-bit A 16x4): lane<16 -> K=k0+{0,1}; lane>=16 -> K=k0+{2,3}
  const int am   = m0 + (lane & 15);
  const int koff = (lane >> 4) * 2;             // 0 or 2
  const int ab   = am / HWPIX;
  const int arem = am - ab * HWPIX;
  const float* xb = x + (size_t)ab * DIMC * HWPIX + arem;

  // B fragment mapping (32-bit B 4x16, mirror of A): row K=k0+koff+{0,1}, col n0+(lane&15)
  const int bn = n0 + (lane & 15);
  const float* wsrc;
  int wld, bcol;
  if (n0 < 128) { wsrc = w_qk; wld = 128; bcol = bn; }
  else          { wsrc = w_v;  wld = 64;  bcol = bn - 128; }

  v8f c = {};
  #pragma unroll
  for (int k0 = 0; k0 < DIMC; k0 += 4) {
    v2f a, b;
    a.x = xb[(size_t)(k0 + koff)     * HWPIX];
    a.y = xb[(size_t)(k0 + koff + 1) * HWPIX];
    b.x = wsrc[(k0 + koff)     * wld + bcol];
    b.y = wsrc[(k0 + koff + 1) * wld + bcol];
    c = __builtin_amdgcn_wmma_f32_16x16x4_f32(false, a, false, b,
                                              (short)0, c, false, false);
  }

  const float sc = (n0 < 64) ? SCALE : 1.0f;    // fold q scaling here
  #pragma unroll
  for (int g = 0; g < 8; ++g) {
    const int m = m0 + g + 8 * (lane >> 4);
    qkv[(size_t)m * QKVW + n0 + (lane & 15)] = c[g] * sc;
  }
}

// ---------------------------------------------------------------------------
// Kernel 2: fused neighborhood attention + superpixel soft-combine.
// One wave32 per (pixel, head). 8 waves / block, grid divides exactly.
// ---------------------------------------------------------------------------
__device__ __forceinline__ float wave_sum(float v) {
  #pragma unroll
  for (int o = 16; o > 0; o >>= 1) v += __shfl_xor(v, o, 32);
  return v;
}
__device__ __forceinline__ float wave_max(float v) {
  #pragma unroll
  for (int o = 16; o > 0; o >>= 1) v = fmaxf(v, __shfl_xor(v, o, 32));
  return v;
}

__global__ void __launch_bounds__(256)
attn_core_kernel(const float* __restrict__ qkv,
                 const float* __restrict__ sims,
                 float* __restrict__ aout) {
  __shared__ float s_attn[8][52];
  __shared__ float s_wcomb[8][52];

  const int lane = threadIdx.x & 31;
  const int wv   = threadIdx.x >> 5;
  const int gw   = blockIdx.x * 8 + wv;   // global wave id
  const int head = gw & 1;
  const int m    = gw >> 1;               // pixel id, < NPIX by construction

  const int b   = m / HWPIX;
  const int rem = m - b * HWPIX;
  const int i   = rem / WW;
  const int j   = rem - i * WW;

  // clipped 7x7 neighborhood window start
  int ri0 = i - KS / 2; ri0 = ri0 < 0 ? 0 : (ri0 > HH - KS ? HH - KS : ri0);
  int cj0 = j - KS / 2; cj0 = cj0 < 0 ? 0 : (cj0 > WW - KS ? WW - KS : cj0);

  // superpixel cell of this pixel
  const int si = i >> 3;                  // i * 12 / 96
  const int sj = j >> 3;

  const int qbase = head * HD;            // q cols [0,64), k at +64, v at +128
  const float q_l = qkv[(size_t)m * QKVW + qbase + lane];
  __builtin_prefetch(&qkv[(size_t)m * QKVW + 128 + qbase], 0, 1);  // v row of center

  // ---- scores: attn[n] = sum_d q*k over HD (lane = hd) ----
  {
    int n = 0;
    for (int dr = 0; dr < KS; ++dr) {
      const int rowm = (b * HH + (ri0 + dr)) * WW + cj0;
      for (int dc = 0; dc < KS; ++dc, ++n) {
        const float kv = qkv[(size_t)(rowm + dc) * QKVW + 64 + qbase + lane];
        float p = wave_sum(q_l * kv);
        if (lane == 0) s_attn[wv][n] = p;
      }
    }
  }
  __syncthreads();

  // ---- per-superpixel softmax + combine (lane = neighbor; 2 slots) ----
  const int n1 = lane;                    // always < 49
  const int n2 = lane + 32;
  const bool has2 = (n2 < K2);

  const int r1 = ri0 + n1 / KS, c1 = cj0 + n1 % KS;
  const int r2 = ri0 + (has2 ? n2 / KS : 0), c2 = cj0 + (has2 ? n2 % KS : 0);
  const size_t pjb1 = (size_t)((b * HH + r1) * WW + c1) * (SHP * SWP);
  const size_t pjb2 = (size_t)((b * HH + r2) * WW + c2) * (SHP * SWP);
  const size_t pib  = (size_t)((b * HH + i) * WW + j) * (SHP * SWP);

  const float a1 = s_attn[wv][n1];
  const float a2 = has2 ? s_attn[wv][n2] : -1e30f;

  float wc1 = 0.0f, wc2 = 0.0f;
  #pragma unroll
  for (int s = 0; s < 9; ++s) {
    int sr = si - 1 + s / 3; sr = sr < 0 ? 0 : (sr > SHP - 1 ? SHP - 1 : sr);
    int sc = sj - 1 + s % 3; sc = sc < 0 ? 0 : (sc > SWP - 1 ? SWP - 1 : sc);
    const int sp = sr * SWP + sc;
    const float pi_s = sims[pib + sp];

    const float l1 = a1 + __logf(sims[pjb1 + sp] + EPSV);
    const float l2 = has2 ? (a2 + __logf(sims[pjb2 + sp] + EPSV)) : -1e30f;

    const float mx = wave_max(fmaxf(l1, l2));
    const float e1 = __expf(l1 - mx);
    const float e2 = has2 ? __expf(l2 - mx) : 0.0f;
    const float ssum = wave_sum(e1 + e2);
    const float w = pi_s / ssum;
    wc1 += e1 * w;
    wc2 += e2 * w;
  }
  s_wcomb[wv][n1] = wc1;
  if (has2) s_wcomb[wv][n2] = wc2;
  __syncthreads();

  // ---- output: out[hd] = sum_n wcomb[n] * v[nbr][hd]  (lane = hd) ----
  float acc = 0.0f;
  {
    int n = 0;
    for (int dr = 0; dr < KS; ++dr) {
      const int rowm = (b * HH + (ri0 + dr)) * WW + cj0;
      for (int dc = 0; dc < KS; ++dc, ++n) {
        const float wn = s_wcomb[wv][n];
        acc += wn * qkv[(size_t)(rowm + dc) * QKVW + 128 + qbase + lane];
      }
    }
  }
  aout[(size_t)m * DIMC + qbase + lane] = acc;
}

// ---------------------------------------------------------------------------
// Kernel 3: output projection (M=18432, K=64, N=64) + bias, fp32 WMMA,
// scatter-store into (B, 64, H, W).
// ---------------------------------------------------------------------------
__global__ void __launch_bounds__(128)
proj_gemm_kernel(const float* __restrict__ ain,
                 const float* __restrict__ w_proj,
                 const float* __restrict__ b_proj,
                 float* __restrict__ out) {
  const int lane = threadIdx.x & 31;
  const int wv   = threadIdx.x >> 5;      // 0..3 -> N tile
  const int m0   = blockIdx.x * 16;
  const int n0   = wv * 16;

  const int am   = m0 + (lane & 15);
  const int koff = (lane >> 4) * 2;
  const int bn   = n0 + (lane & 15);

  v8f c = {};
  #pragma unroll
  for (int k0 = 0; k0 < DIMC; k0 += 4) {
    v2f a, b;
    a.x = ain[(size_t)am * DIMC + k0 + koff];
    a.y = ain[(size_t)am * DIMC + k0 + koff + 1];
    b.x = w_proj[(k0 + koff)     * DIMC + bn];
    b.y = w_proj[(k0 + koff + 1) * DIMC + bn];
    c = __builtin_amdgcn_wmma_f32_16x16x4_f32(false, a, false, b,
                                              (short)0, c, false, false);
  }

  const float bias = b_proj[bn];
  #pragma unroll
  for (int g = 0; g < 8; ++g) {
    const int m  = m0 + g + 8 * (lane >> 4);
    const int bb = m / HWPIX;
    const int rm = m - bb * HWPIX;
    out[((size_t)bb * DIMC + bn) * HWPIX + rm] = c[g] + bias;
  }
}

// ---------------------------------------------------------------------------
extern "C" void kernel_launch(void* const* d_in, const int* in_sizes, int n_in,
                              void* d_out, int out_size, void* d_ws, size_t ws_size,
                              hipStream_t stream) {
  const float* x      = (const float*)d_in[0];
  const float* sims   = (const float*)d_in[1];
  const float* w_qk   = (const float*)d_in[2];
  const float* w_v    = (const float*)d_in[3];
  const float* w_proj = (const float*)d_in[4];
  const float* b_proj = (const float*)d_in[5];
  float* out = (float*)d_out;

  float* qkv  = (float*)d_ws;                       // NPIX * 192 floats
  float* aout = qkv + (size_t)NPIX * QKVW;          // NPIX * 64 floats

  dim3 g1(NPIX / 16, 3, 1);                         // 1152 x 3, 4 waves/block
  qkv_gemm_kernel<<<g1, 128, 0, stream>>>(x, w_qk, w_v, qkv);

  const int nwaves = NPIX * HEADS;                  // 36864
  attn_core_kernel<<<nwaves / 8, 256, 0, stream>>>(qkv, sims, aout);

  proj_gemm_kernel<<<NPIX / 16, 128, 0, stream>>>(aout, w_proj, b_proj, out);
}